// Similarity_Loss_v2_46334107189861
// MI455X (gfx1250) — compile-verified
//
#include <hip/hip_runtime.h>
#include <hip/hip_bf16.h>

typedef __attribute__((ext_vector_type(2))) float v2f;
typedef __attribute__((ext_vector_type(8))) float v8f;

#define NPATCH 20
#define BATCH  4096
#define DIM    1024
#define WPB    4          // waves per block (128 threads)

// One wave computes the 20x20 Gram of z_list[:, b, :] via v_wmma_f32_16x16x4_f32,
// then all 60 cosine terms for batch element b, and writes their sum to ws[b].
__global__ __launch_bounds__(WPB * 32)
void gram_cos_kernel(const float* __restrict__ z, float* __restrict__ ws)
{
    __shared__ float gram[WPB][32 * 32];
    __shared__ float rs[WPB][32];

    const int tid  = threadIdx.x;
    const int lane = tid & 31;
    const int w    = tid >> 5;
    const int b    = blockIdx.x * WPB + w;

    const int c = lane & 15;   // M (A) / N (B) index within 16
    const int h = lane >> 4;   // selects K pair {2h, 2h+1}

    // rows 0..15 always valid; rows 16..31: only 16..19 valid -> clamp + zero
    const int   r0     = c;
    const int   r1v    = 16 + c;
    const float valid1 = (r1v < NPATCH) ? 1.0f : 0.0f;
    const int   r1     = (r1v < NPATCH) ? r1v : 0;

    // z_list layout [N, B, D]: row n of batch b starts at (n*B + b)*D
    const size_t base0 = ((size_t)r0 * BATCH + b) * DIM;
    const size_t base1 = ((size_t)r1 * BATCH + b) * DIM;

    v8f g00 = {}; v8f g01 = {}; v8f g11 = {};

    #pragma unroll 4
    for (int k0 = 0; k0 < DIM; k0 += 4) {
        const int koff = k0 + 2 * h;                   // this lane's K pair
        v2f a0 = *(const v2f*)(z + base0 + koff);      // rows 0..15 operand
        v2f a1 = *(const v2f*)(z + base1 + koff);      // rows 16..19 (padded)
        a1 = a1 * valid1;
        // Gram: fp32 A-layout == fp32 B-layout, so the same regs feed A and B.
        g00 = __builtin_amdgcn_wmma_f32_16x16x4_f32(false, a0, false, a0,
                                                    (short)0, g00, false, false);
        g01 = __builtin_amdgcn_wmma_f32_16x16x4_f32(false, a0, false, a1,
                                                    (short)0, g01, false, false);
        g11 = __builtin_amdgcn_wmma_f32_16x16x4_f32(false, a1, false, a1,
                                                    (short)0, g11, false, false);
    }

    // C/D layout: lane (c + 16*hh) vgpr r holds D[M = r + 8*hh][N = c]
    const int rowh = h * 8;
    #pragma unroll
    for (int r = 0; r < 8; ++r) {
        const int m = r + rowh;
        gram[w][m * 32 + c]              = g00[r];   // G[m][c]
        gram[w][m * 32 + 16 + c]         = g01[r];   // G[m][16+c]
        gram[w][(16 + c) * 32 + m]       = g01[r];   // symmetry: G[16+c][m]
        gram[w][(16 + m) * 32 + 16 + c]  = g11[r];   // G[16+m][16+c]
    }
    __syncthreads();

    if (lane < NPATCH) {
        float s = 0.0f;
        #pragma unroll
        for (int mm = 0; mm < NPATCH; ++mm) s += gram[w][lane * 32 + mm];
        rs[w][lane] = s;   // rowsum[n] = 20 * dot(x_n, z_avg)
    }
    __syncthreads();

    float total = 0.0f;
    #pragma unroll
    for (int n = 0; n < NPATCH; ++n) total += rs[w][n];   // = 400*||z_avg||^2

    const float EPSV     = 1e-8f;
    const float navg     = fmaxf(sqrtf(total) * (1.0f / NPATCH), EPSV);
    const float inv_navg = 1.0f / navg;
    const float inv_n    = 1.0f / (float)NPATCH;

    float part = 0.0f;
    // 20 original patches: one per lane
    if (lane < NPATCH) {
        const float dot = rs[w][lane] * inv_n;
        const float nn  = fmaxf(sqrtf(gram[w][lane * 33]), EPSV);
        part = dot / nn * inv_navg;
    }
    // 40 extras: pair tables packed into immediates (PA 2b/entry, PB 3b/entry)
    const unsigned PA = (1u << 8) | (1u << 10) | (1u << 12) |
                        (2u << 14) | (2u << 16) | (3u << 18);
    const unsigned PB = 1u | (2u << 3) | (3u << 6) | (4u << 9) | (2u << 12) |
                        (3u << 15) | (4u << 18) | (3u << 21) | (4u << 24) | (4u << 27);
    for (int e = lane; e < 40; e += 32) {
        const int i  = e / 10;
        const int pr = e - i * 10;
        const int p  = (int)((PA >> (2 * pr)) & 3u) * 4 + i;
        const int q  = (int)((PB >> (3 * pr)) & 7u) * 4 + i;
        const float dot = 0.5f * (rs[w][p] + rs[w][q]) * inv_n;
        const float n2  = 0.25f * (gram[w][p * 33] +
                                   2.0f * gram[w][p * 32 + q] +
                                   gram[w][q * 33]);
        part += dot / fmaxf(sqrtf(n2), EPSV) * inv_navg;
    }

    // wave32 reduction of the 60 cosine terms
    #pragma unroll
    for (int off = 16; off > 0; off >>= 1)
        part += __shfl_down(part, off, 32);
    if (lane == 0) ws[b] = part;
}

// Deterministic fixed-order final reduction over the 4096 per-batch sums.
__global__ __launch_bounds__(256)
void finalize_kernel(const float* __restrict__ ws, float* __restrict__ out)
{
    __shared__ float s[256];
    float acc = 0.0f;
    for (int i = threadIdx.x; i < BATCH; i += 256) acc += ws[i];
    s[threadIdx.x] = acc;
    __syncthreads();
    for (int st = 128; st > 0; st >>= 1) {
        if (threadIdx.x < st) s[threadIdx.x] += s[threadIdx.x + st];
        __syncthreads();
    }
    if (threadIdx.x == 0) {
        const float z_sim = s[0] / (60.0f * (float)BATCH);
        out[0] = -z_sim;
        out[1] = z_sim;
    }
}

extern "C" void kernel_launch(void* const* d_in, const int* in_sizes, int n_in,
                              void* d_out, int out_size, void* d_ws, size_t ws_size,
                              hipStream_t stream)
{
    (void)in_sizes; (void)n_in; (void)out_size; (void)ws_size;
    const float* z   = (const float*)d_in[0];   // z_list [20, 4096, 1024] fp32
    float*       ws  = (float*)d_ws;            // 4096 floats, fully overwritten
    float*       out = (float*)d_out;           // 2 floats: (-z_sim, z_sim)

    gram_cos_kernel<<<BATCH / WPB, WPB * 32, 0, stream>>>(z, ws);
    finalize_kernel<<<1, 256, 0, stream>>>(ws, out);
}